// PointNextDecoder_34737695490785
// MI455X (gfx1250) — compile-verified
//
#include <hip/hip_runtime.h>

typedef __attribute__((ext_vector_type(2))) float v2f;
typedef __attribute__((ext_vector_type(8))) float v8f;

// ---------------------------------------------------------------------------
// 3-NN inverse-distance interpolation:
//   p1 [B,N1,3], p2 [B,N2,3], f2 [B,C,N2]  ->  out [B,C,N1]
// one thread per (b, n1)
// ---------------------------------------------------------------------------
__global__ void knn_interp_kernel(const float* __restrict__ p1,
                                  const float* __restrict__ p2,
                                  const float* __restrict__ f2,
                                  float* __restrict__ out,
                                  int N1, int N2, int C)
{
    const int n = blockIdx.x * blockDim.x + threadIdx.x;
    const int b = blockIdx.y;
    if (n >= N1) return;

    const float* pp = p1 + ((size_t)b * N1 + n) * 3;
    const float px = pp[0], py = pp[1], pz = pp[2];

    float d0 = 3.4e38f, d1 = 3.4e38f, d2 = 3.4e38f;
    int   i0 = 0, i1 = 0, i2 = 0;

    const float* q = p2 + (size_t)b * N2 * 3;
    for (int j = 0; j < N2; ++j) {
        const float dx = px - q[3 * j + 0];
        const float dy = py - q[3 * j + 1];
        const float dz = pz - q[3 * j + 2];
        const float d  = dx * dx + dy * dy + dz * dz;
        if (d < d0)      { d2 = d1; i2 = i1; d1 = d0; i1 = i0; d0 = d; i0 = j; }
        else if (d < d1) { d2 = d1; i2 = i1; d1 = d;  i1 = j; }
        else if (d < d2) { d2 = d;  i2 = j; }
    }

    float w0 = 1.0f / (d0 + 1e-8f);
    float w1 = 1.0f / (d1 + 1e-8f);
    float w2 = 1.0f / (d2 + 1e-8f);
    const float ws = 1.0f / (w0 + w1 + w2);
    w0 *= ws; w1 *= ws; w2 *= ws;

    const float* fb = f2 + (size_t)b * C * N2;
    float*       ob = out + (size_t)b * C * N1;
    for (int c = 0; c < C; ++c) {
        const float* row = fb + (size_t)c * N2;
        ob[(size_t)c * N1 + n] = w0 * row[i0] + w1 * row[i1] + w2 * row[i2];
    }
}

// ---------------------------------------------------------------------------
// Fused 1x1-conv (GEMM) + BN + ReLU using V_WMMA_F32_16X16X4_F32.
//   out[b,o,n] = relu( g[o] * sum_c W[o,c] * X[b,c,n] + bias[o] )
// X is the virtual concat of X1 [B,C1,N] (skip) and X2 [B,C2,N] (interp).
//
// One wave (block = 32) computes a 16(M) x 64(N) strip with 4 independent
// accumulators: the A fragment is loaded once per K-step and reused by 4
// WMMAs, and the 4 acc chains interleave to hide WMMA latency.
// Grid: (N/64, cout/16, B). All dims divide exactly -> no predication,
// EXEC is all-ones at every WMMA (ISA requirement).
//
// A-matrix 16x4 f32 layout (ISA 7.12.2): lanes 0-15 -> M=lane, VGPR0/1 = K0/K1;
// lanes 16-31 -> M=lane-16, VGPR0/1 = K2/K3. B 4x16 is the mirrored column
// layout. C/D: VGPR r holds M = r + 8*(lane>=16), N = lane&15.
// ---------------------------------------------------------------------------
__global__ void __launch_bounds__(32)
wmma_gemm_bn_relu(const float* __restrict__ W,
                  const float* __restrict__ X1, int C1,
                  const float* __restrict__ X2, int C2,
                  const float* __restrict__ g,
                  const float* __restrict__ bias,
                  float* __restrict__ out,
                  int N, int cout)
{
    const int lane = threadIdx.x & 31;
    const int lh   = lane & 15;    // row (A) / column (B) within tile
    const int hi   = lane >> 4;    // 0: K pair {0,1}, 1: K pair {2,3}

    const int n0  = blockIdx.x * 64;
    const int m0  = blockIdx.y * 16;
    const int b   = blockIdx.z;
    const int cin = C1 + C2;

    // A: weight row (m0+lh), starting at K offset 2*hi
    const float* wrow = W + (size_t)(m0 + lh) * cin + 2 * hi;
    // B: feature column base (n0+lh); 4 sub-tiles at +0,+16,+32,+48
    const float* x1b = (C1 > 0) ? (X1 + (size_t)b * C1 * N + (n0 + lh)) : nullptr;
    const float* x2b = X2 + (size_t)b * C2 * N + (n0 + lh);

    v8f acc0 = {}, acc1 = {}, acc2 = {}, acc3 = {};
    for (int k = 0; k < cin; k += 4) {
        v2f a;
        a.x = wrow[k + 0];
        a.y = wrow[k + 1];
        const int kk = k + 2 * hi;              // lane's K base; same side as k (C1 % 4 == 0)
        const float* src = (k < C1) ? (x1b + (size_t)kk * N)
                                    : (x2b + (size_t)(kk - C1) * N);
        v2f b0, b1, b2, b3;
        b0.x = src[0];       b0.y = src[N];
        b1.x = src[16];      b1.y = src[N + 16];
        b2.x = src[32];      b2.y = src[N + 32];
        b3.x = src[48];      b3.y = src[N + 48];

        acc0 = __builtin_amdgcn_wmma_f32_16x16x4_f32(false, a, false, b0, (short)0, acc0, false, false);
        acc1 = __builtin_amdgcn_wmma_f32_16x16x4_f32(false, a, false, b1, (short)0, acc1, false, false);
        acc2 = __builtin_amdgcn_wmma_f32_16x16x4_f32(false, a, false, b2, (short)0, acc2, false, false);
        acc3 = __builtin_amdgcn_wmma_f32_16x16x4_f32(false, a, false, b3, (short)0, acc3, false, false);
    }

    float* ob = out + (size_t)b * cout * N + (n0 + lh);
#pragma unroll
    for (int r = 0; r < 8; ++r) {
        const int o   = m0 + r + 8 * hi;
        const float sc = g[o];
        const float sh = bias[o];
        float* orow = ob + (size_t)o * N;
        float v0 = acc0[r] * sc + sh;
        float v1 = acc1[r] * sc + sh;
        float v2 = acc2[r] * sc + sh;
        float v3 = acc3[r] * sc + sh;
        orow[0]  = v0 > 0.0f ? v0 : 0.0f;
        orow[16] = v1 > 0.0f ? v1 : 0.0f;
        orow[32] = v2 > 0.0f ? v2 : 0.0f;
        orow[48] = v3 > 0.0f ? v3 : 0.0f;
    }
}

// ---------------------------------------------------------------------------
// Host side: 4 decoder stages, deepest first.
// ---------------------------------------------------------------------------
extern "C" void kernel_launch(void* const* d_in, const int* in_sizes, int n_in,
                              void* d_out, int out_size, void* d_ws, size_t ws_size,
                              hipStream_t stream)
{
    (void)in_sizes; (void)n_in; (void)out_size; (void)ws_size;

    const float* p[5];
    const float* f[5];
    for (int i = 0; i < 5; ++i) {
        p[i] = (const float*)d_in[2 * i + 0];
        f[i] = (const float*)d_in[2 * i + 1];
    }
    const float *wa[4], *ga[4], *ba[4], *wb[4], *gb[4], *bb[4];
    for (int s = 0; s < 4; ++s) {
        wa[s] = (const float*)d_in[10 + 6 * s + 0];
        ga[s] = (const float*)d_in[10 + 6 * s + 1];
        ba[s] = (const float*)d_in[10 + 6 * s + 2];
        wb[s] = (const float*)d_in[10 + 6 * s + 3];
        gb[s] = (const float*)d_in[10 + 6 * s + 4];
        bb[s] = (const float*)d_in[10 + 6 * s + 5];
    }

    const int B = 8;
    const int Ns[5]  = {8192, 2048, 512, 128, 32};
    const int CHs[5] = {32, 64, 128, 256, 512};

    // workspace slots (floats), reused across the 4 sequential stages
    float* ws        = (float*)d_ws;
    float* interpBuf = ws;                       // max 8*64*8192   = 4,194,304
    float* yBuf      = ws + 4194304;             // max 8*32*8192   = 2,097,152
    float* outBuf    = ws + 4194304 + 2097152;   // max 8*64*2048   = 1,048,576

    const float* prevOut = f[4];   // deepest features, C = 512
    int prevC = CHs[4];

    for (int s = 0; s < 4; ++s) {
        const int lvl  = 3 - s;
        const int N1   = Ns[lvl];
        const int N2   = Ns[lvl + 1];
        const int C1   = CHs[lvl];   // skip channels
        const int C2   = prevC;      // interpolated channels (prev stage out)
        const int cout = CHs[lvl];

        // 1) 3-NN interpolate prevOut [B,C2,N2] -> interpBuf [B,C2,N1]
        dim3 gi(N1 / 128, B);
        knn_interp_kernel<<<gi, 128, 0, stream>>>(p[lvl], p[lvl + 1], prevOut,
                                                  interpBuf, N1, N2, C2);

        // 2) conv_a: [cout, C1+C2] @ concat(f[lvl], interp) -> yBuf, +BN+ReLU
        dim3 gg(N1 / 64, cout / 16, B);
        wmma_gemm_bn_relu<<<gg, 32, 0, stream>>>(wa[s], f[lvl], C1,
                                                 interpBuf, C2,
                                                 ga[s], ba[s], yBuf, N1, cout);

        // 3) conv_b: [cout, cout] @ yBuf -> stage output, +BN+ReLU
        float* dst = (s == 3) ? (float*)d_out : outBuf;
        wmma_gemm_bn_relu<<<gg, 32, 0, stream>>>(wb[s], nullptr, 0,
                                                 yBuf, cout,
                                                 gb[s], bb[s], dst, N1, cout);

        prevOut = dst;
        prevC   = cout;
    }
}